// GNNFraudDetector_15547781612037
// MI455X (gfx1250) — compile-verified
//
#include <hip/hip_runtime.h>
#include <hip/hip_bf16.h>

typedef __attribute__((ext_vector_type(2))) float v2f;
typedef __attribute__((ext_vector_type(8))) float v8f;

// ---------------------------------------------------------------------------
// Degree kernels: deg[i] = 1 (self loop) + #incoming edges; then dis = rsqrt
// ---------------------------------------------------------------------------
__global__ void gcn_deg_init(float* __restrict__ deg, int n) {
    int i = blockIdx.x * blockDim.x + threadIdx.x;
    if (i < n) deg[i] = 1.0f;
}

__global__ void gcn_deg_count(const long long* __restrict__ dst,
                              float* __restrict__ deg, int e) {
    int i = blockIdx.x * blockDim.x + threadIdx.x;
    if (i < e) atomicAdd(&deg[(size_t)dst[i]], 1.0f);
}

__global__ void gcn_deg_rsqrt(float* __restrict__ deg, int n) {
    int i = blockIdx.x * blockDim.x + threadIdx.x;
    if (i < n) deg[i] = rsqrtf(deg[i]);   // deg buffer becomes dis = deg^{-1/2}
}

// ---------------------------------------------------------------------------
// 128-wide fused GEMM: H = act(A) @ W  (K = N = 128), one 16-row tile/block,
// 8 waves each owning a 16-col tile via V_WMMA_F32_16X16X4_F32.
// W (64KB) and the 16x128 A tile (stride-130 padded) are staged in LDS;
// ReLU is applied during A staging (template, no in-loop branches).
// Epilogue writes H (raw) and AGG = H*dis[row]^2 + bias (self-loop + bias).
// ---------------------------------------------------------------------------
template <bool RELU>
__global__ __launch_bounds__(256)
void gcn_gemm128_wmma(const float* __restrict__ A,    // [M x 128]
                      const float* __restrict__ W,    // [128 x 128]
                      const float* __restrict__ bias, // [128]
                      const float* __restrict__ dis,  // [M]
                      float* __restrict__ H,          // [M x 128]
                      float* __restrict__ AGG,        // [M x 128]
                      int M) {
    __shared__ float Wlds[128 * 128];   // [k][col], natural layout
    __shared__ float Alds[16 * 130];    // [row][k], padded stride

    const int tid  = threadIdx.x;
    const int wave = tid >> 5;
    const int lane = tid & 31;
    const int half = lane >> 4;         // K sub-pair selector
    const int lr   = lane & 15;
    const int rowBase = blockIdx.x * 16;
    const int col  = wave * 16 + lr;

    // Stage W: 4096 float4, coalesced b128 loads -> ds_store_b128
    {
        const float4* Wg = (const float4*)W;
        float4*       Wl = (float4*)Wlds;
        #pragma unroll
        for (int j = 0; j < 16; ++j) Wl[tid + 256 * j] = Wg[tid + 256 * j];
    }
    // Stage A tile (16 x 128) with fused ReLU
    {
        #pragma unroll
        for (int j = 0; j < 8; ++j) {
            const int i = tid + 256 * j;        // 0..2047
            const int r = i >> 7, k = i & 127;
            float a = A[(size_t)(rowBase + r) * 128 + k];
            if (RELU) a = fmaxf(a, 0.0f);
            Alds[r * 130 + k] = a;
        }
    }
    __syncthreads();

    v8f c = {};
    #pragma unroll
    for (int k0 = 0; k0 < 128; k0 += 4) {
        const int ka = k0 + 2 * half;
        const v2f a = *(const v2f*)&Alds[lr * 130 + ka];   // ds_load_b64
        v2f b;
        b.x = Wlds[(ka + 0) * 128 + col];
        b.y = Wlds[(ka + 1) * 128 + col];
        c = __builtin_amdgcn_wmma_f32_16x16x4_f32(
                false, a, false, b, (short)0, c, false, false);
    }

    const float bcol = bias[col];
    #pragma unroll
    for (int v = 0; v < 8; ++v) {
        const int row = rowBase + v + 8 * half;
        const float h  = c[v];
        const float ds = dis[row];
        H  [(size_t)row * 128 + col] = h;
        AGG[(size_t)row * 128 + col] = h * (ds * ds) + bcol;
    }
}

// ---------------------------------------------------------------------------
// Narrow output GEMM (N <= 16, here N = 2): one wave per 16-row tile.
// Branch-free masking: clamp column index, scale loads by 0/1. ReLU on A.
// ---------------------------------------------------------------------------
__global__ __launch_bounds__(32)
void gcn_gemm_narrow_wmma(const float* __restrict__ A,    // [M x 128]
                          const float* __restrict__ W,    // [128 x N]
                          const float* __restrict__ bias, // [N]
                          const float* __restrict__ dis,  // [M]
                          float* __restrict__ H,          // [M x N]
                          float* __restrict__ AGG,        // [M x N]
                          int M, int N) {
    const int lane = threadIdx.x & 31;
    const int half = lane >> 4;
    const int lr   = lane & 15;
    const int rowBase = blockIdx.x * 16;
    const bool valid = lr < N;
    const int  colc  = valid ? lr : 0;
    const float fm   = valid ? 1.0f : 0.0f;

    const float* ap = A + (size_t)(rowBase + lr) * 128;

    v8f c = {};
    #pragma unroll
    for (int k0 = 0; k0 < 128; k0 += 4) {
        const int ka = k0 + 2 * half;
        v2f a;
        a.x = fmaxf(ap[ka + 0], 0.0f);
        a.y = fmaxf(ap[ka + 1], 0.0f);
        v2f b;
        b.x = W[(size_t)(ka + 0) * N + colc] * fm;
        b.y = W[(size_t)(ka + 1) * N + colc] * fm;
        c = __builtin_amdgcn_wmma_f32_16x16x4_f32(
                false, a, false, b, (short)0, c, false, false);
    }

    if (valid) {
        const float bcol = bias[lr];
        #pragma unroll
        for (int v = 0; v < 8; ++v) {
            const int row = rowBase + v + 8 * half;
            const float h  = c[v];
            const float ds = dis[row];
            H  [(size_t)row * N + lr] = h;
            AGG[(size_t)row * N + lr] = h * (ds * ds) + bcol;
        }
    }
}

// ---------------------------------------------------------------------------
// Edge scatter, 128 channels: one wave per edge, float4 per lane.
// AGG[dst] += H[src] * dis[src]*dis[dst]
// ---------------------------------------------------------------------------
__global__ void gcn_edge_scatter128(const float* __restrict__ H,
                                    float* __restrict__ AGG,
                                    const long long* __restrict__ src,
                                    const long long* __restrict__ dst,
                                    const float* __restrict__ dis,
                                    int e) {
    const int warp = (int)((blockIdx.x * (size_t)blockDim.x + threadIdx.x) >> 5);
    const int lane = threadIdx.x & 31;
    if (warp >= e) return;
    const long long s = src[warp];
    const long long d = dst[warp];
    const float norm = dis[s] * dis[d];
    const float4 h = ((const float4*)(H + (size_t)s * 128))[lane];
    float* ap = AGG + (size_t)d * 128 + lane * 4;
    atomicAdd(ap + 0, h.x * norm);
    atomicAdd(ap + 1, h.y * norm);
    atomicAdd(ap + 2, h.z * norm);
    atomicAdd(ap + 3, h.w * norm);
}

// Edge scatter, 2 channels (output layer): one thread per edge.
__global__ void gcn_edge_scatter2(const float* __restrict__ H,
                                  float* __restrict__ OUT,
                                  const long long* __restrict__ src,
                                  const long long* __restrict__ dst,
                                  const float* __restrict__ dis,
                                  int e) {
    const int i = blockIdx.x * blockDim.x + threadIdx.x;
    if (i >= e) return;
    const long long s = src[i];
    const long long d = dst[i];
    const float norm = dis[s] * dis[d];
    atomicAdd(OUT + (size_t)d * 2 + 0, H[(size_t)s * 2 + 0] * norm);
    atomicAdd(OUT + (size_t)d * 2 + 1, H[(size_t)s * 2 + 1] * norm);
}

// ---------------------------------------------------------------------------
// Host launcher
// ---------------------------------------------------------------------------
extern "C" void kernel_launch(void* const* d_in, const int* in_sizes, int n_in,
                              void* d_out, int out_size, void* d_ws, size_t ws_size,
                              hipStream_t stream) {
    const float*     x   = (const float*)d_in[0];
    const long long* ei  = (const long long*)d_in[1];   // int64 [2, E]
    const float*     W1  = (const float*)d_in[2];
    const float*     b1  = (const float*)d_in[3];
    const float*     W2  = (const float*)d_in[4];
    const float*     b2  = (const float*)d_in[5];
    const float*     W3  = (const float*)d_in[6];
    const float*     b3  = (const float*)d_in[7];

    const int CH  = 128;
    const int OC  = 2;
    const int N   = in_sizes[0] / CH;     // 100000
    const int E   = in_sizes[1] / 2;      // 1600000
    const long long* src = ei;
    const long long* dst = ei + E;

    // Workspace layout (floats)
    float* ws  = (float*)d_ws;
    float* dis = ws;                                 // [N]
    size_t off = ((size_t)N + 255) & ~(size_t)255;
    float* B0  = ws + off;                           // [N*128]  H scratch
    float* B1  = B0 + (size_t)N * CH;                // [N*128]  AGG / features
    float* B2  = B1 + (size_t)N * CH;                // [N*128]  AGG / features
    float* H3  = B2 + (size_t)N * CH;                // [N*2]    raw XW3
    float* out = (float*)d_out;                      // [N*2]

    const int nRowTiles = (N + 15) / 16;             // 6250 (N % 16 == 0)

    // Degrees -> dis
    gcn_deg_init <<<(N + 255) / 256, 256, 0, stream>>>(dis, N);
    gcn_deg_count<<<(E + 255) / 256, 256, 0, stream>>>(dst, dis, E);
    gcn_deg_rsqrt<<<(N + 255) / 256, 256, 0, stream>>>(dis, N);

    const int scatterBlocks = (int)(((size_t)E * 32 + 255) / 256);

    // Layer 1: H1 = x @ W1 ; AGG1 = self+bias ; scatter edges
    gcn_gemm128_wmma<false><<<nRowTiles, 256, 0, stream>>>(x, W1, b1, dis,
                                                           B0, B1, N);
    gcn_edge_scatter128<<<scatterBlocks, 256, 0, stream>>>(B0, B1, src, dst, dis, E);

    // Layer 2: H2 = relu(AGG1) @ W2 ; AGG2 = self+bias ; scatter
    gcn_gemm128_wmma<true><<<nRowTiles, 256, 0, stream>>>(B1, W2, b2, dis,
                                                          B0, B2, N);
    gcn_edge_scatter128<<<scatterBlocks, 256, 0, stream>>>(B0, B2, src, dst, dis, E);

    // Layer 3: H3 = relu(AGG2) @ W3 ; out = self+bias ; scatter (2 ch)
    gcn_gemm_narrow_wmma<<<nRowTiles, 32, 0, stream>>>(B2, W3, b3, dis,
                                                       H3, out, N, OC);
    gcn_edge_scatter2<<<(E + 255) / 256, 256, 0, stream>>>(H3, out, src, dst, dis, E);
}